// ContrastiveLoss_25116968747114
// MI455X (gfx1250) — compile-verified
//
#include <hip/hip_runtime.h>
#include <hip/hip_bf16.h>

// Contrastive (NT-Xent) loss, fused Gram + logsumexp, f32 WMMA path.
// B=4096, D=128, N=8192, T=0.5. Logits bounded in [-2,2] -> fixed-max lse.
// Round 2: software-pipelined B-fragment loads (double buffer, half-tile
// granularity) so 16 global_load_b64 stay in flight behind each 16-WMMA chain.

typedef __attribute__((ext_vector_type(2))) float v2f;
typedef __attribute__((ext_vector_type(8))) float v8f;

#define BATCH 4096
#define DIM   128
#define NTOT  8192          // 2*BATCH
#define INV_T 2.0f          // 1/temperature
#define LMAX  2.0f          // fixed logsumexp max: |cos|<=1 -> |logit|<=2

// ---------------------------------------------------------------------------
// Kernel 1: row-wise L2 normalize z = [z1;z2] -> zn. One wave per row.
// ---------------------------------------------------------------------------
__global__ void nt_normalize_kernel(const float* __restrict__ z1,
                                    const float* __restrict__ z2,
                                    float* __restrict__ zn) {
  const int row  = blockIdx.x * 8 + (threadIdx.x >> 5);   // 8 waves / block
  const int lane = threadIdx.x & 31;
  const float* src = (row < BATCH) ? (z1 + (size_t)row * DIM)
                                   : (z2 + (size_t)(row - BATCH) * DIM);
  float4 v = ((const float4*)src)[lane];                  // 32 lanes * 4 = 128
  float ss = v.x * v.x + v.y * v.y + v.z * v.z + v.w * v.w;
#pragma unroll
  for (int m = 1; m < 32; m <<= 1) ss += __shfl_xor(ss, m, 32);
  const float inv = 1.0f / fmaxf(sqrtf(ss), 1e-8f);       // torch eps clamp
  float4 o; o.x = v.x * inv; o.y = v.y * inv; o.z = v.z * inv; o.w = v.w * inv;
  ((float4*)(zn + (size_t)row * DIM))[lane] = o;
}

// ---------------------------------------------------------------------------
// Kernel 2: pos[i] = pos[i+B] = (sim[i,i+B]+sim[i+B,i])/T = 4*dot(zn_i,zn_{i+B})
// ---------------------------------------------------------------------------
__global__ void nt_pos_kernel(const float* __restrict__ zn,
                              float* __restrict__ pos) {
  const int i    = blockIdx.x * 8 + (threadIdx.x >> 5);
  const int lane = threadIdx.x & 31;
  float4 a = ((const float4*)(zn + (size_t)i * DIM))[lane];
  float4 b = ((const float4*)(zn + (size_t)(i + BATCH) * DIM))[lane];
  float d = a.x * b.x + a.y * b.y + a.z * b.z + a.w * b.w;
#pragma unroll
  for (int m = 1; m < 32; m <<= 1) d += __shfl_xor(d, m, 32);
  if (lane == 0) {
    const float p = 2.0f * INV_T * d;
    pos[i] = p;
    pos[i + BATCH] = p;
  }
}

// ---------------------------------------------------------------------------
// Kernel 3: fused sim-tile (f32 WMMA 16x16x4) + diagonal-masked sum(exp(x-2)).
// Grid: NTOT/16 = 512 blocks of 256 threads (8 waves).
// Each wave: same 16 rows, its own 64 column tiles (of 512 total).
// A fragments (16 rows x K=128) live in registers across the column loop.
// B fragments double-buffered at half-tile (16-WMMA) granularity so loads
// overlap the WMMA chains instead of serializing on s_wait_loadcnt 0.
// ---------------------------------------------------------------------------
__global__ void __launch_bounds__(256)
nt_sim_lse_kernel(const float* __restrict__ zn,
                  const float* __restrict__ pos,
                  float* __restrict__ partial) {
  const int rowBase = blockIdx.x * 16;
  const int wave = threadIdx.x >> 5;     // 0..7
  const int lane = threadIdx.x & 31;
  const int m    = lane & 15;            // A: row-in-tile; B: col-in-tile
  const int h    = lane >> 4;            // K-pair selector (see 16x4 layout)

  // A fragment cache: lane holds zn[rowBase+m][4k + 2h .. +1] for k = 0..31
  const float* arow = zn + (size_t)(rowBase + m) * DIM + 2 * h;
  v2f a[32];
#pragma unroll
  for (int k = 0; k < 32; ++k) a[k] = *(const v2f*)(arow + 4 * k);

  float sum8[8];
#pragma unroll
  for (int v = 0; v < 8; ++v) sum8[v] = 0.0f;

  const int ctBeg = wave * 64;
  const int ctEnd = ctBeg + 64;

  // Lane-relative B pointer for a given column tile ct.
  const float* brow = zn + (size_t)(ctBeg * 16 + m) * DIM + 2 * h;

  v2f b0[16], b1[16];
  // Prime the pipeline: half 0 of the first tile.
#pragma unroll
  for (int k = 0; k < 16; ++k) b0[k] = *(const v2f*)(brow + 4 * k);

  for (int ct = ctBeg; ct < ctEnd; ++ct) {
    const int colBase = ct * 16;
    const float* brow_next = brow + 16 * DIM;   // next tile (stray read of
                                                // last+1 tile lands in pos[])
    v8f c = {};

    // Issue loads for half 1 of this tile while computing half 0.
#pragma unroll
    for (int k = 0; k < 16; ++k) b1[k] = *(const v2f*)(brow + 64 + 4 * k);
#pragma unroll
    for (int k = 0; k < 16; ++k) {
      c = __builtin_amdgcn_wmma_f32_16x16x4_f32(
          false, a[k], false, b0[k], (short)0, c, false, false);
    }

    // Issue loads for half 0 of the NEXT tile while computing half 1.
#pragma unroll
    for (int k = 0; k < 16; ++k) b0[k] = *(const v2f*)(brow_next + 4 * k);
#pragma unroll
    for (int k = 0; k < 16; ++k) {
      c = __builtin_amdgcn_wmma_f32_16x16x4_f32(
          false, a[16 + k], false, b1[k], (short)0, c, false, false);
    }

    // D layout: vgpr v, lane l -> M = v + 8*(l>>4), N = l&15
    const int col = colBase + m;
#pragma unroll
    for (int v = 0; v < 8; ++v) {
      const int r = rowBase + v + 8 * h;
      const float x = INV_T * c[v] - LMAX;       // logit - fixed max
      if (r != col) sum8[v] += __expf(x);        // mask diagonal only
    }

    brow = brow_next;
  }

  // Combine the 16 column-residue lanes in each half (rows fixed per half).
#pragma unroll
  for (int msk = 1; msk <= 8; msk <<= 1)
#pragma unroll
    for (int v = 0; v < 8; ++v) sum8[v] += __shfl_xor(sum8[v], msk, 32);

  // Cross-wave combine via LDS: lsum[wave][row-in-tile]
  __shared__ float lsum[8][16];
  if (m == 0) {
#pragma unroll
    for (int v = 0; v < 8; ++v) lsum[wave][8 * h + v] = sum8[v];
  }
  __syncthreads();

  if (wave == 0 && lane < 16) {
    float tot = 0.0f;
#pragma unroll
    for (int w = 0; w < 8; ++w) tot += lsum[w][lane];
    const float lse_neg = LMAX + __logf(tot);           // <= 2 + log(8191)
    const float p = pos[rowBase + lane];
    const float mx = fmaxf(p, lse_neg);
    const float lse = mx + __logf(__expf(p - mx) + __expf(lse_neg - mx));
    float contrib = lse - p;
#pragma unroll
    for (int msk = 1; msk <= 8; msk <<= 1)
      contrib += __shfl_xor(contrib, msk, 32);
    if (lane == 0) partial[blockIdx.x] = contrib;       // deterministic
  }
}

// ---------------------------------------------------------------------------
// Kernel 4: reduce 512 per-rowtile partials -> loss = total / N^2.
// ---------------------------------------------------------------------------
__global__ void nt_finalize_kernel(const float* __restrict__ partial,
                                   float* __restrict__ out) {
  __shared__ float s[256];
  const int t = threadIdx.x;
  s[t] = partial[t] + partial[t + 256];
  __syncthreads();
  for (int off = 128; off > 0; off >>= 1) {
    if (t < off) s[t] += s[t + off];
    __syncthreads();
  }
  if (t == 0) out[0] = s[0] * (1.0f / ((float)NTOT * (float)NTOT));
}

// ---------------------------------------------------------------------------
extern "C" void kernel_launch(void* const* d_in, const int* in_sizes, int n_in,
                              void* d_out, int out_size, void* d_ws, size_t ws_size,
                              hipStream_t stream) {
  (void)in_sizes; (void)n_in; (void)out_size; (void)ws_size;
  const float* z1 = (const float*)d_in[0];
  const float* z2 = (const float*)d_in[1];
  float* out = (float*)d_out;

  // Workspace layout: zn (N*D f32 = 4MB) | pos (N f32) | partial (512 f32)
  float* zn      = (float*)d_ws;
  float* pos     = zn + (size_t)NTOT * DIM;
  float* partial = pos + NTOT;

  nt_normalize_kernel<<<NTOT / 8, 256, 0, stream>>>(z1, z2, zn);
  nt_pos_kernel<<<BATCH / 8, 256, 0, stream>>>(zn, pos);
  nt_sim_lse_kernel<<<NTOT / 16, 256, 0, stream>>>(zn, pos, partial);
  nt_finalize_kernel<<<1, 256, 0, stream>>>(partial, out);
}